// cnn_lut_77180562309371
// MI455X (gfx1250) — compile-verified
//
#include <hip/hip_runtime.h>
#include <hip/hip_bf16.h>

typedef __attribute__((ext_vector_type(2))) float v2f;
typedef __attribute__((ext_vector_type(8))) float v8f;

#define DEVINL __device__ __forceinline__

// D = A(16x4 f32) x B(4x16 f32) + C(16x16 f32), native fp32 WMMA on gfx1250.
DEVINL v8f wmma4(float a0, float a1, float b0, float b1, v8f c) {
  v2f A; A.x = a0; A.y = a1;
  v2f Bv; Bv.x = b0; Bv.y = b1;
  return __builtin_amdgcn_wmma_f32_16x16x4_f32(false, A, false, Bv, (short)0, c,
                                               false, false);
}

// Select one of 8 wave-uniform values by k (0..7): cndmask tree, no scratch.
DEVINL unsigned sel8(unsigned k, unsigned v0, unsigned v1, unsigned v2, unsigned v3,
                     unsigned v4, unsigned v5, unsigned v6, unsigned v7) {
  unsigned a = (k & 1) ? v1 : v0;
  unsigned b = (k & 1) ? v3 : v2;
  unsigned c = (k & 1) ? v5 : v4;
  unsigned d = (k & 1) ? v7 : v6;
  unsigned e = (k & 2) ? b : a;
  unsigned f = (k & 2) ? d : c;
  return (k & 4) ? f : e;
}

// ---------------------------------------------------------------------------
// Precompute 1: sign-pattern -> codebook-index decoder tables.
// For each H[15,16]: idx(mask) = argmax_h sum_k (mask_k ? +1 : -1) * H[k][h].
// 4 tables x 32768 bytes, written to ws[0 .. 131072).
// ---------------------------------------------------------------------------
__global__ __launch_bounds__(256) void build_sign(const float* __restrict__ H3,
                                                  const float* __restrict__ H4,
                                                  const float* __restrict__ H5,
                                                  const float* __restrict__ H2,
                                                  unsigned char* __restrict__ out) {
  int t = blockIdx.x * blockDim.x + threadIdx.x;
  if (t >= 4 * 32768) return;
  int mat = t >> 15;
  unsigned mask = t & 0x7FFFu;
  const float* H = (mat == 0) ? H3 : (mat == 1) ? H4 : (mat == 2) ? H5 : H2;
  float z[16];
#pragma unroll
  for (int h = 0; h < 16; ++h) z[h] = 0.f;
#pragma unroll
  for (int k = 0; k < 15; ++k) {
    float s = ((mask >> k) & 1u) ? 1.f : -1.f;
#pragma unroll
    for (int h = 0; h < 16; ++h) z[h] += s * H[k * 16 + h];
  }
  int best = 0;
  float bv = z[0];
#pragma unroll
  for (int h = 1; h < 16; ++h) {
    if (z[h] > bv) { bv = z[h]; best = h; }  // first-max, matches jnp.argmax
  }
  out[t] = (unsigned char)best;
}

// ---------------------------------------------------------------------------
// Precompute 2: fuse LUT3/4/5 [nk,C,16,O] -> g[C][16][nk*O] (nk outer, O inner)
// so stage-2 gathers read contiguous (o, o+1) float2 pairs.
// ---------------------------------------------------------------------------
__global__ __launch_bounds__(256) void build_g(const float* __restrict__ L3,
                                               const float* __restrict__ L4,
                                               const float* __restrict__ L5,
                                               float* __restrict__ g3,
                                               float* __restrict__ g4,
                                               float* __restrict__ g5) {
  int t = blockIdx.x * blockDim.x + threadIdx.x;
  const int N3 = 32 * 16 * 96, N4 = 32 * 16 * 80, N5 = 32 * 16 * 64;
  if (t < N3) {
    int o = t % 96, k = (t / 96) & 15, c = t / (96 * 16);
    int j = o / 6, oo = o % 6;
    g3[t] = L3[((j * 32 + c) * 16 + k) * 6 + oo];
  } else if (t < N3 + N4) {
    int u = t - N3;
    int o = u % 80, k = (u / 80) & 15, c = u / (80 * 16);
    int j = o / 5, oo = o % 5;
    g4[u] = L4[((j * 32 + c) * 16 + k) * 5 + oo];
  } else if (t < N3 + N4 + N5) {
    int u = t - N3 - N4;
    int o = u & 63, k = (u >> 6) & 15, c = u >> 10;
    int j = o >> 2, oo = o & 3;
    g5[u] = L5[((j * 32 + c) * 16 + k) * 4 + oo];
  }
}

// ---------------------------------------------------------------------------
// Main fused kernel: 256 threads (8 wave32), 32 rows per wave (two 16-row
// WMMA tiles sharing B operands). All decode tables resident in LDS.
// ---------------------------------------------------------------------------
__global__ __launch_bounds__(256) void fused_main(
    const int* __restrict__ x, const float* __restrict__ lenLUT,
    const float* __restrict__ ipdLUT, const float* __restrict__ S3,
    const float* __restrict__ T3, const float* __restrict__ S4,
    const float* __restrict__ T4, const float* __restrict__ S5,
    const float* __restrict__ T5, const float* __restrict__ S2,
    const float* __restrict__ T2, const float* __restrict__ LUT2,
    const unsigned char* __restrict__ wsSign, const float* __restrict__ g3,
    const float* __restrict__ g4, const float* __restrict__ g5,
    float* __restrict__ out, int B) {
  __shared__ __attribute__((aligned(16))) unsigned char sLUT[4 * 32768];
  __shared__ __attribute__((aligned(16))) unsigned char sIdx3[8][32][32];
  __shared__ __attribute__((aligned(16))) unsigned char sIdx4[8][32][32];
  __shared__ __attribute__((aligned(16))) unsigned char sIdx5[8][32][32];
  __shared__ __attribute__((aligned(16))) unsigned char sIdx2[8][32][16];

  // Cooperative fill of the 128KB decoder tables (L2 -> LDS once per WG).
  {
    const uint4* src = (const uint4*)wsSign;
    uint4* dst = (uint4*)sLUT;
    for (int i = threadIdx.x; i < (4 * 32768) / 16; i += 256) dst[i] = src[i];
  }
  __syncthreads();

  const int lane = (int)(threadIdx.x & 31u);
  const int wave = (int)(threadIdx.x >> 5u);
  const int m = lane & 15;            // A-layout row (lanes l and l+16 share row)
  const int half = (lane >> 4) & 1;   // K phase: 0 -> K,K+1 ; 1 -> K+2,K+3
  const int n = lane & 15;            // B/C column
  const long rowBase = (long)blockIdx.x * 256 + (long)wave * 32;
  long r0 = rowBase + m;        if (r0 > (long)B - 1) r0 = (long)B - 1;
  long r1 = rowBase + m + 16;   if (r1 > (long)B - 1) r1 = (long)B - 1;

  unsigned char* idx3p = &sIdx3[wave][0][0];
  unsigned char* idx4p = &sIdx4[wave][0][0];
  unsigned char* idx5p = &sIdx5[wave][0][0];
  unsigned char* idx2p = &sIdx2[wave][0][0];

  // One codebook matmul + decode for both 16-row tiles of this wave.
  auto stage1_mat = [&](int c, const float* a0, const float* a1,
                        const float* __restrict__ S, const float* __restrict__ T,
                        int signOff, unsigned char* idxp) {
    float b[8];
#pragma unroll
    for (int j = 0; j < 4; ++j) {
      int d0 = 4 * j + 2 * half;
      b[2 * j]     = (n < 15) ? S[(c * 16 + d0) * 15 + n] : 0.f;
      b[2 * j + 1] = (n < 15) ? S[(c * 16 + d0 + 1) * 15 + n] : 0.f;
    }
    float tc = (n < 15) ? T[c * 15 + n] : 3.0e38f;
    v8f acc0 = {0.f, 0.f, 0.f, 0.f, 0.f, 0.f, 0.f, 0.f};
    v8f acc1 = {0.f, 0.f, 0.f, 0.f, 0.f, 0.f, 0.f, 0.f};
#pragma unroll
    for (int j = 0; j < 4; ++j) {
      acc0 = wmma4(a0[2 * j], a0[2 * j + 1], b[2 * j], b[2 * j + 1], acc0);
      acc1 = wmma4(a1[2 * j], a1[2 * j + 1], b[2 * j], b[2 * j + 1], acc1);
    }
    unsigned bal0[8], bal1[8];
#pragma unroll
    for (int i = 0; i < 8; ++i) {
      bal0[i] = (unsigned)__ballot(acc0[i] > tc);
      bal1[i] = (unsigned)__ballot(acc1[i] > tc);
    }
    unsigned k7 = (unsigned)(lane & 7);
    unsigned va = sel8(k7, bal0[0], bal0[1], bal0[2], bal0[3], bal0[4], bal0[5],
                       bal0[6], bal0[7]);
    unsigned vb = sel8(k7, bal1[0], bal1[1], bal1[2], bal1[3], bal1[4], bal1[5],
                       bal1[6], bal1[7]);
    unsigned v = (lane & 16) ? vb : va;
    unsigned mask = ((lane & 8) ? (v >> 16) : v) & 0x7FFFu;
    idxp[lane * 32 + c] = sLUT[signOff + (int)mask];  // lane r owns row r
  };

  // ---------------- Stage 1: emb -> 3x (WMMA + decode) over c=0..31 --------
  for (int l = 0; l < 8; ++l) {
    const int xa0 = x[(r0 * 8 + l) * 2 + 0];
    const int xa1 = x[(r0 * 8 + l) * 2 + 1];
    const int xb0 = x[(r1 * 8 + l) * 2 + 0];
    const int xb1 = x[(r1 * 8 + l) * 2 + 1];
    const float* pa0 = lenLUT + (long)xa0 * 64;
    const float* pa1 = ipdLUT + (long)xa1 * 64;
    const float* pb0 = lenLUT + (long)xb0 * 64;
    const float* pb1 = ipdLUT + (long)xb1 * 64;
#pragma unroll
    for (int cc = 0; cc < 4; ++cc) {
      const int c = l * 4 + cc;
      float a0[8], a1[8];  // A-layout tiles: emb rows, K=d
#pragma unroll
      for (int j = 0; j < 4; ++j) {
        int e0 = cc * 16 + 4 * j + 2 * half;
        float2 LA = *(const float2*)(pa0 + e0);
        float2 IA = *(const float2*)(pa1 + e0);
        float2 LB = *(const float2*)(pb0 + e0);
        float2 IB = *(const float2*)(pb1 + e0);
        a0[2 * j] = LA.x + IA.x;  a0[2 * j + 1] = LA.y + IA.y;
        a1[2 * j] = LB.x + IB.x;  a1[2 * j + 1] = LB.y + IB.y;
      }
      stage1_mat(c, a0, a1, S3, T3, 0 * 32768, idx3p);
      stage1_mat(c, a0, a1, S4, T4, 1 * 32768, idx4p);
      stage1_mat(c, a0, a1, S5, T5, 2 * 32768, idx5p);
    }
  }

  // In-wave LDS store->load handoff (DS ops are wave-ordered; wait + barrier
  // the compiler from reordering across the alias).
  asm volatile("s_wait_dscnt 0x0" ::: "memory");

  // ---------------- Stage 2: gather xc tiles, WMMA vs S2, decode -----------
  auto stage2_range = [&](const float* __restrict__ g, int OC, int OBASE,
                          const unsigned char* idxArr, int c2lo, int c2hi) {
    unsigned wa[8], wb[8];  // 32 stage-1 idx bytes for each of this lane's rows
    const unsigned* qp = (const unsigned*)idxArr;
#pragma unroll
    for (int q = 0; q < 8; ++q) {
      wa[q] = qp[m * 8 + q];
      wb[q] = qp[(m + 16) * 8 + q];
    }
    for (int c2 = c2lo; c2 < c2hi; ++c2) {
      float a0[8], a1[8];
#pragma unroll
      for (int j = 0; j < 4; ++j) {
        int o0 = c2 * 16 + 4 * j + 2 * half - OBASE;
        float sx0 = 0.f, sy0 = 0.f, sx1 = 0.f, sy1 = 0.f;
#pragma unroll
        for (int cc = 0; cc < 32; ++cc) {
          unsigned ia = (wa[cc >> 2] >> (8 * (cc & 3))) & 255u;
          unsigned ib = (wb[cc >> 2] >> (8 * (cc & 3))) & 255u;
          float2 va = *(const float2*)(g + ((cc * 16 + (int)ia) * OC + o0));
          float2 vb = *(const float2*)(g + ((cc * 16 + (int)ib) * OC + o0));
          sx0 += va.x; sy0 += va.y;
          sx1 += vb.x; sy1 += vb.y;
        }
        a0[2 * j] = sx0; a0[2 * j + 1] = sy0;
        a1[2 * j] = sx1; a1[2 * j + 1] = sy1;
      }
      float b[8];
#pragma unroll
      for (int j = 0; j < 4; ++j) {
        int d0 = 4 * j + 2 * half;
        b[2 * j]     = (n < 15) ? S2[(c2 * 16 + d0) * 15 + n] : 0.f;
        b[2 * j + 1] = (n < 15) ? S2[(c2 * 16 + d0 + 1) * 15 + n] : 0.f;
      }
      float tc = (n < 15) ? T2[c2 * 15 + n] : 3.0e38f;
      v8f acc0 = {0.f, 0.f, 0.f, 0.f, 0.f, 0.f, 0.f, 0.f};
      v8f acc1 = {0.f, 0.f, 0.f, 0.f, 0.f, 0.f, 0.f, 0.f};
#pragma unroll
      for (int j = 0; j < 4; ++j) {
        acc0 = wmma4(a0[2 * j], a0[2 * j + 1], b[2 * j], b[2 * j + 1], acc0);
        acc1 = wmma4(a1[2 * j], a1[2 * j + 1], b[2 * j], b[2 * j + 1], acc1);
      }
      unsigned bal0[8], bal1[8];
#pragma unroll
      for (int i = 0; i < 8; ++i) {
        bal0[i] = (unsigned)__ballot(acc0[i] > tc);
        bal1[i] = (unsigned)__ballot(acc1[i] > tc);
      }
      unsigned k7 = (unsigned)(lane & 7);
      unsigned va8 = sel8(k7, bal0[0], bal0[1], bal0[2], bal0[3], bal0[4],
                          bal0[5], bal0[6], bal0[7]);
      unsigned vb8 = sel8(k7, bal1[0], bal1[1], bal1[2], bal1[3], bal1[4],
                          bal1[5], bal1[6], bal1[7]);
      unsigned v = (lane & 16) ? vb8 : va8;
      unsigned mask = ((lane & 8) ? (v >> 16) : v) & 0x7FFFu;
      idx2p[lane * 16 + c2] = sLUT[3 * 32768 + (int)mask];
    }
  };
  stage2_range(g3, 96, 0, idx3p, 0, 6);     // xc cols [  0, 96)
  stage2_range(g4, 80, 96, idx4p, 6, 11);   // xc cols [ 96,176)
  stage2_range(g5, 64, 176, idx5p, 11, 15); // xc cols [176,240)

  asm volatile("s_wait_dscnt 0x0" ::: "memory");

  // ---------------- Final: LUT2 gather-sum + log_softmax(128) --------------
  for (int r = 0; r < 32; ++r) {
    long row = rowBase + r;
    unsigned iw[4];
#pragma unroll
    for (int q = 0; q < 4; ++q) iw[q] = ((const unsigned*)idx2p)[r * 4 + q];
    float4 acc = {0.f, 0.f, 0.f, 0.f};
#pragma unroll
    for (int c2 = 0; c2 < 15; ++c2) {
      unsigned idx = (iw[c2 >> 2] >> (8 * (c2 & 3))) & 255u;
      float4 v = *(const float4*)(LUT2 + ((long)(c2 * 16 + (int)idx) * 128 + lane * 4));
      acc.x += v.x; acc.y += v.y; acc.z += v.z; acc.w += v.w;
    }
    float mx = fmaxf(fmaxf(acc.x, acc.y), fmaxf(acc.z, acc.w));
#pragma unroll
    for (int d = 16; d >= 1; d >>= 1) mx = fmaxf(mx, __shfl_xor(mx, d, 32));
    float se = expf(acc.x - mx) + expf(acc.y - mx) + expf(acc.z - mx) +
               expf(acc.w - mx);
#pragma unroll
    for (int d = 16; d >= 1; d >>= 1) se += __shfl_xor(se, d, 32);
    float ls = mx + logf(se);
    if (row < (long)B) {
      float4 o4;
      o4.x = acc.x - ls; o4.y = acc.y - ls; o4.z = acc.z - ls; o4.w = acc.w - ls;
      *(float4*)(out + row * 128 + lane * 4) = o4;
    }
  }
}

// ---------------------------------------------------------------------------
// ws layout (needs >= 622592 bytes):
//   [0,131072)          4x 32768-byte sign-decoder tables
//   [131072,327680)     g3 [32][16][96] f32
//   [327680,491520)     g4 [32][16][80] f32
//   [491520,622592)     g5 [32][16][64] f32
// ---------------------------------------------------------------------------
extern "C" void kernel_launch(void* const* d_in, const int* in_sizes, int n_in,
                              void* d_out, int out_size, void* d_ws,
                              size_t ws_size, hipStream_t stream) {
  const int* x = (const int*)d_in[0];
  const float* lenLUT = (const float*)d_in[1];
  const float* ipdLUT = (const float*)d_in[2];
  const float* S3 = (const float*)d_in[3];
  const float* T3 = (const float*)d_in[4];
  const float* H3 = (const float*)d_in[5];
  const float* L3 = (const float*)d_in[6];
  const float* S4 = (const float*)d_in[7];
  const float* T4 = (const float*)d_in[8];
  const float* H4 = (const float*)d_in[9];
  const float* L4 = (const float*)d_in[10];
  const float* S5 = (const float*)d_in[11];
  const float* T5 = (const float*)d_in[12];
  const float* H5 = (const float*)d_in[13];
  const float* L5 = (const float*)d_in[14];
  const float* S2 = (const float*)d_in[15];
  const float* T2 = (const float*)d_in[16];
  const float* H2 = (const float*)d_in[17];
  const float* L2 = (const float*)d_in[18];
  float* out = (float*)d_out;
  const int B = in_sizes[0] / 16;  // x is [B,8,2]

  unsigned char* wsSign = (unsigned char*)d_ws;
  float* g3 = (float*)((char*)d_ws + 131072);
  float* g4 = (float*)((char*)d_ws + 327680);
  float* g5 = (float*)((char*)d_ws + 491520);

  build_sign<<<(4 * 32768 + 255) / 256, 256, 0, stream>>>(H3, H4, H5, H2, wsSign);
  build_g<<<(32 * 16 * (96 + 80 + 64) + 255) / 256, 256, 0, stream>>>(L3, L4, L5,
                                                                      g3, g4, g5);
  const int nblk = (B + 255) / 256;
  fused_main<<<nblk, 256, 0, stream>>>(x, lenLUT, ipdLUT, S3, T3, S4, T4, S5, T5,
                                       S2, T2, L2, wsSign, g3, g4, g5, out, B);
}